// EncoderGCN_69045894250778
// MI455X (gfx1250) — compile-verified
//
#include <hip/hip_runtime.h>
#include <hip/hip_bf16.h>

typedef __attribute__((ext_vector_type(2))) float v2f;
typedef __attribute__((ext_vector_type(8))) float v8f;

#define DIMF 128
#define NGRAPH 128
#define OUTSTRIDE 384

// ---------------------------------------------------------------- utilities
__global__ void k_zero_f32(float* __restrict__ p, long n) {
    long i = (long)blockIdx.x * blockDim.x + threadIdx.x;
    if (i < n) p[i] = 0.0f;
}

// ---------------------------------------------------------------- degree / norm
__global__ void k_degree(const int* __restrict__ dst, float* __restrict__ deg, int E) {
    int e = blockIdx.x * blockDim.x + threadIdx.x;
    if (e < E) atomicAdd(&deg[dst[e]], 1.0f);
}

// dinv[i] = rsqrt(deg[i] + 1)   (+1 accounts for the self loop; deg+1 >= 1 always)
__global__ void k_dinv(float* __restrict__ deg_dinv, int N) {
    int i = blockIdx.x * blockDim.x + threadIdx.x;
    if (i < N) deg_dinv[i] = rsqrtf(deg_dinv[i] + 1.0f);
}

__global__ void k_edge_norm(const int* __restrict__ src, const int* __restrict__ dst,
                            const float* __restrict__ dinv, float* __restrict__ norm, int E) {
    int e = blockIdx.x * blockDim.x + threadIdx.x;
    if (e < E) norm[e] = dinv[src[e]] * dinv[dst[e]];
}

// ---------------------------------------------------------------- GEMM (WMMA f32 16x16x4)
// C[M,128] = A[M,128] @ W[128,128].  One wave -> one 16x16 tile, 8 waves/block
// cover N=128, grid.x covers M/16 tiles (M is a multiple of 16 here: 50000).
__global__ void __launch_bounds__(256)
k_gemm_wmma_f32(const float* __restrict__ A, const float* __restrict__ W,
                float* __restrict__ C, int M) {
    const int wave = threadIdx.x >> 5;      // 0..7 -> N tile
    const int lane = threadIdx.x & 31;
    const int lo16 = lane & 15;
    const int khi  = lane >> 4;             // 0 or 1
    const int m0   = blockIdx.x * 16;
    const int n0   = wave * 16;
    const long arow = (long)(m0 + lo16) * DIMF;

    v8f acc = {};
#pragma unroll
    for (int k0 = 0; k0 < DIMF; k0 += 4) {
        const int ka = k0 + 2 * khi;
        v2f a, b;
        a.x = A[arow + ka + 0];
        a.y = A[arow + ka + 1];
        b.x = W[(long)(ka + 0) * DIMF + n0 + lo16];
        b.y = W[(long)(ka + 1) * DIMF + n0 + lo16];
        acc = __builtin_amdgcn_wmma_f32_16x16x4_f32(
            /*neg_a=*/false, a, /*neg_b=*/false, b,
            /*c_mod=*/(short)0, acc, /*reuse_a=*/false, /*reuse_b=*/false);
    }
    // D layout: VGPR r -> lanes 0-15: M=r, N=lane; lanes 16-31: M=r+8, N=lane-16
    const int n     = n0 + lo16;
    const int mbase = m0 + khi * 8;
#pragma unroll
    for (int r = 0; r < 8; ++r)
        C[(long)(mbase + r) * DIMF + n] = acc[r];
}

// ---------------------------------------------------------------- aggregation
// Self loops initialize the aggregate (every node has exactly one): agg = hW * dinv^2
__global__ void k_selfloop_init(const float* __restrict__ hw, const float* __restrict__ dinv,
                                float* __restrict__ agg, long total) {
    long i = (long)blockIdx.x * blockDim.x + threadIdx.x;
    if (i >= total) return;
    float dv = dinv[i >> 7];          // node = i / 128
    agg[i] = hw[i] * dv * dv;
}

// 32 lanes per edge, float4 per lane: one 512B row read per edge, f32 atomics (L2-resident)
__global__ void __launch_bounds__(256)
k_edge_scatter(const float* __restrict__ hw, const int* __restrict__ src,
               const int* __restrict__ dst, const float* __restrict__ norm,
               float* __restrict__ agg, int E) {
    int e = blockIdx.x * 8 + (threadIdx.x >> 5);
    if (e >= E) return;
    int lane = threadIdx.x & 31;
    int s = src[e], d = dst[e];
    float nm = norm[e];
    const float4* hp = (const float4*)(hw + (long)s * DIMF);
    float4 v = hp[lane];
    float* ap = agg + (long)d * DIMF + lane * 4;
    atomicAdd(ap + 0, v.x * nm);
    atomicAdd(ap + 1, v.y * nm);
    atomicAdd(ap + 2, v.z * nm);
    atomicAdd(ap + 3, v.w * nm);
}

// ---------------------------------------------------------------- BN (fused bias+ReLU+stats)
// thread f handles feature f; block strides rows; writes relu(agg+b) back in place.
__global__ void __launch_bounds__(128)
k_bias_relu_stats(float* __restrict__ agg, const float* __restrict__ bias,
                  float* __restrict__ sums, int N) {
    int f = threadIdx.x;
    float b = bias[f];
    float s = 0.0f, s2 = 0.0f;
    for (int i = blockIdx.x; i < N; i += gridDim.x) {
        long idx = (long)i * DIMF + f;
        float v = fmaxf(agg[idx] + b, 0.0f);
        agg[idx] = v;
        s += v;
        s2 += v * v;
    }
    atomicAdd(&sums[f], s);
    atomicAdd(&sums[DIMF + f], s2);
}

__global__ void k_bn_finalize(const float* __restrict__ sums, const float* __restrict__ g,
                              const float* __restrict__ be, float* __restrict__ ss, float invN) {
    int f = threadIdx.x;
    float mu  = sums[f] * invN;
    float var = sums[DIMF + f] * invN - mu * mu;
    float rs  = rsqrtf(var + 1e-5f);
    float sc  = rs * g[f];
    ss[f]        = sc;
    ss[DIMF + f] = be[f] - mu * sc;
}

// BN apply fused with graph pooling (segment_sum over batch) into d_out[g, layer*128+f]
__global__ void __launch_bounds__(128)
k_bn_apply_pool(const float* __restrict__ agg, const float* __restrict__ ss,
                const int* __restrict__ batch, float* __restrict__ hout,
                float* __restrict__ out, int layer, int N) {
    int f = threadIdx.x;
    float sc = ss[f], sh = ss[DIMF + f];
    for (int i = blockIdx.x; i < N; i += gridDim.x) {
        long idx = (long)i * DIMF + f;
        float v = agg[idx] * sc + sh;
        hout[idx] = v;
        atomicAdd(&out[(long)batch[i] * OUTSTRIDE + layer * DIMF + f], v);
    }
}

// ---------------------------------------------------------------- host
static inline size_t align256(size_t x) { return (x + 255) & ~(size_t)255; }

extern "C" void kernel_launch(void* const* d_in, const int* in_sizes, int n_in,
                              void* d_out, int out_size, void* d_ws, size_t ws_size,
                              hipStream_t stream) {
    const float* x     = (const float*)d_in[0];
    const int*   ei    = (const int*)d_in[1];
    const int*   batch = (const int*)d_in[2];

    const int N = in_sizes[0] / DIMF;       // 50000
    const int E = in_sizes[1] / 2;          // 1600000
    const int* src = ei;
    const int* dst = ei + E;

    // workspace carve-up
    char* ws = (char*)d_ws;
    size_t off = 0;
    float* dinv = (float*)(ws + off); off += align256((size_t)N * 4);
    float* norm = (float*)(ws + off); off += align256((size_t)E * 4);
    float* bufA = (float*)(ws + off); off += align256((size_t)N * DIMF * 4);
    float* bufB = (float*)(ws + off); off += align256((size_t)N * DIMF * 4);
    float* sums = (float*)(ws + off); off += align256(2 * DIMF * 4);
    float* ss   = (float*)(ws + off); off += align256(2 * DIMF * 4);
    (void)ws_size;

    const long NF = (long)N * DIMF;

    // --- normalization coefficients
    k_zero_f32<<<(N + 255) / 256, 256, 0, stream>>>(dinv, N);
    k_degree<<<(E + 255) / 256, 256, 0, stream>>>(dst, dinv, E);
    k_dinv<<<(N + 255) / 256, 256, 0, stream>>>(dinv, N);
    k_edge_norm<<<(E + 255) / 256, 256, 0, stream>>>(src, dst, dinv, norm, E);

    // --- zero the pooled output (harness poisons it)
    k_zero_f32<<<((long)out_size + 255) / 256, 256, 0, stream>>>((float*)d_out, out_size);

    const float* hin = x;
    float* p0 = bufA;   // GEMM output (hW), then reused as h_next
    float* p1 = bufB;   // aggregate buffer

    for (int l = 0; l < 3; ++l) {
        const float* W  = (const float*)d_in[3 + 4 * l];
        const float* b  = (const float*)d_in[4 + 4 * l];
        const float* g  = (const float*)d_in[5 + 4 * l];
        const float* be = (const float*)d_in[6 + 4 * l];

        // hW = hin @ W   (WMMA f32)
        k_gemm_wmma_f32<<<N / 16, 256, 0, stream>>>(hin, W, p0, N);

        // agg = selfloop contribution (initializes p1), then edge scatter-add
        k_selfloop_init<<<(NF + 255) / 256, 256, 0, stream>>>(p0, dinv, p1, NF);
        k_edge_scatter<<<(E + 7) / 8, 256, 0, stream>>>(p0, src, dst, norm, p1, E);

        // fused bias + ReLU + BN statistics (in place on p1)
        k_zero_f32<<<1, 256, 0, stream>>>(sums, 2 * DIMF);
        k_bias_relu_stats<<<512, 128, 0, stream>>>(p1, b, sums, N);
        k_bn_finalize<<<1, 128, 0, stream>>>(sums, g, be, ss, 1.0f / (float)N);

        // BN apply -> h_next (into p0, now dead) fused with graph pooling
        k_bn_apply_pool<<<512, 128, 0, stream>>>(p1, ss, batch, p0, (float*)d_out, l, N);

        hin = p0;
        float* t = p0; p0 = p1; p1 = t;
    }
}